// NSE_69415261438113
// MI455X (gfx1250) — compile-verified
//
#include <hip/hip_runtime.h>
#include <cmath>

// MI455X (gfx1250) implementation notes:
//  - The model is a 512-step sequential scan (~6 GFLOP, ~25MB live state):
//    latency-bound, fully L2-resident. HBM (23.3 TB/s) is irrelevant; the
//    binding resources are per-WGP L2 bandwidth and cross-step serialization.
//  - Phase A: the only scan-independent GEMM (x@Wr+br for all 256 steps,
//    8192x512x128) runs chip-wide with V_WMMA_F32_16X16X4_F32.
//  - Phase B: reader LSTM scan on ONE WGP with Ur (256KB) staged in the 320KB
//    LDS; 2048 f32-WMMAs/step, LDS-bandwidth balanced.
//  - Phase C: writer scan on 32 WGPs with the 4MB attention memory DISTRIBUTED
//    across their LDS (128KB slice each) -> all mem streaming is LDS traffic
//    instead of 4GB of single-WGP L2 traffic. Cross-WGP sync = 2 spin
//    barriers/step (generation counter in workspace, memset to 0 per launch so
//    graph replays are deterministic; 1 block/WGP -> co-residency guaranteed).
//    The small per-step LSTM is recomputed redundantly per block (deterministic)
//    to avoid any broadcast. m_rt is reduced with GLOBAL_ATOMIC_ADD_F32 into a
//    parity-double-buffered accumulator.
//  - f32 WMMA keeps full precision (LSTM recurrence over 256 steps punishes
//    fp16/bf16, and we are nowhere near FLOP-bound).

typedef float v2f __attribute__((ext_vector_type(2)));
typedef float v8f __attribute__((ext_vector_type(8)));

#define B_ 32
#define L_ 256
#define D_ 128
#define G_ 512           // 4*D
#define NBLK_ 32         // writer blocks (1 per WGP)
#define LSL_ 8           // L_/NBLK_  l-values per writer block

static __device__ __forceinline__ v8f wmma_f32_16x16x4(v2f a, v2f b, v8f c) {
  return __builtin_amdgcn_wmma_f32_16x16x4_f32(false, a, false, b, (short)0, c,
                                               false, false);
}

static __device__ __forceinline__ float hard_sigmoid(float x) {
  return fminf(fmaxf(0.2f * x + 0.5f, 0.0f), 1.0f);
}

// Generation-counter global spin barrier (all NBLK_ blocks participate).
static __device__ __forceinline__ void gbarrier(unsigned* cnt, unsigned* gen,
                                                unsigned& lgen) {
  __syncthreads();
  if (threadIdx.x == 0) {
    __builtin_amdgcn_fence(__ATOMIC_RELEASE, "agent");
    const unsigned target = ++lgen;
    const unsigned old = __hip_atomic_fetch_add(cnt, 1u, __ATOMIC_RELAXED,
                                                __HIP_MEMORY_SCOPE_AGENT);
    if (old == (unsigned)(NBLK_ - 1)) {
      __hip_atomic_store(cnt, 0u, __ATOMIC_RELAXED, __HIP_MEMORY_SCOPE_AGENT);
      __hip_atomic_fetch_add(gen, 1u, __ATOMIC_RELEASE,
                             __HIP_MEMORY_SCOPE_AGENT);
    } else {
      while (__hip_atomic_load(gen, __ATOMIC_RELAXED,
                               __HIP_MEMORY_SCOPE_AGENT) < target)
        __builtin_amdgcn_s_sleep(2);
    }
    __builtin_amdgcn_fence(__ATOMIC_ACQUIRE, "agent");
  }
  __syncthreads();
}

// ---------------------------------------------------------------------------
// Phase A: XW[t*32+b][n] = sum_k x[b][t][k]*Wr[k][n] + br[n]  (chip-wide WMMA)
// ---------------------------------------------------------------------------
__global__ void precompute_xw_kernel(const float* __restrict__ x,
                                     const float* __restrict__ Wr,
                                     const float* __restrict__ br,
                                     float* __restrict__ XW) {
  const int wid  = threadIdx.x >> 5;
  const int lane = threadIdx.x & 31;
  const int lm   = lane & 15;
  const int hi   = lane >> 4;
  const int tile = blockIdx.x * (blockDim.x >> 5) + wid;  // 0..16383
  const int mt = tile >> 5;
  const int nt = tile & 31;
  const int n  = nt * 16 + lm;
  const int R0 = mt * 16;

  const int Ra = R0 + lm;               // row R = t*32 + b -> x[b][t][:]
  const float* arow = x + ((size_t)(Ra & 31) * L_ + (Ra >> 5)) * D_;

  v8f acc;
  const float bias = br[n];
#pragma unroll
  for (int vr = 0; vr < 8; ++vr) acc[vr] = bias;

#pragma unroll 4
  for (int k = 0; k < D_; k += 4) {
    const int kb = k + 2 * hi;
    v2f a = *(const v2f*)(arow + kb);
    v2f b;
    b.x = Wr[(size_t)kb * G_ + n];
    b.y = Wr[(size_t)(kb + 1) * G_ + n];
    acc = wmma_f32_16x16x4(a, b, acc);
  }

#pragma unroll
  for (int vr = 0; vr < 8; ++vr)
    XW[(size_t)(R0 + vr + 8 * hi) * G_ + n] = acc[vr];
}

// ---------------------------------------------------------------------------
// Phase B: reader LSTM scan. 1 WG x 512 threads; Ur + h + c in LDS (288KB).
// ---------------------------------------------------------------------------
__global__ void reader_scan_kernel(const float* __restrict__ XW,
                                   const float* __restrict__ Ur,
                                   float* __restrict__ oseq) {
  extern __shared__ float sm[];
  float* Ur_s = sm;                  // 128*512
  float* h_s  = Ur_s + D_ * G_;      // 32*128
  float* c_s  = h_s + B_ * D_;       // 32*128

  const int tid  = threadIdx.x;
  const int wid  = tid >> 5;
  const int lane = tid & 31;
  const int lm   = lane & 15;
  const int hi   = lane >> 4;
  const int m0   = (wid & 1) * 16;
  const int j0   = (wid >> 1) * 16;

  for (int i = tid; i < (D_ * G_) / 4; i += blockDim.x)
    ((float4*)Ur_s)[i] = ((const float4*)Ur)[i];
  for (int i = tid; i < B_ * D_; i += blockDim.x) { h_s[i] = 0.f; c_s[i] = 0.f; }
  __syncthreads();

  for (int t = 0; t < L_; ++t) {
    const float* xwt = XW + (size_t)t * B_ * G_;
    v8f acc[4];
#pragma unroll
    for (int g = 0; g < 4; ++g)
#pragma unroll
      for (int vr = 0; vr < 8; ++vr)
        acc[g][vr] = xwt[(m0 + vr + 8 * hi) * G_ + g * D_ + j0 + lm];

#pragma unroll 2
    for (int k = 0; k < D_; k += 4) {
      const int kb = k + 2 * hi;
      v2f a = *(const v2f*)&h_s[(m0 + lm) * D_ + kb];
#pragma unroll
      for (int g = 0; g < 4; ++g) {
        const int col = g * D_ + j0 + lm;
        v2f b;
        b.x = Ur_s[kb * G_ + col];
        b.y = Ur_s[(kb + 1) * G_ + col];
        acc[g] = wmma_f32_16x16x4(a, b, acc[g]);
      }
    }
    __syncthreads();

    float* ot = oseq + (size_t)t * B_ * D_;
#pragma unroll
    for (int vr = 0; vr < 8; ++vr) {
      const int row = m0 + vr + 8 * hi;
      const int col = j0 + lm;
      const float iv = hard_sigmoid(acc[0][vr]);
      const float fv = hard_sigmoid(acc[1][vr]);
      const float gv = tanhf(acc[2][vr]);
      const float ov = hard_sigmoid(acc[3][vr]);
      const float cn = fv * c_s[row * D_ + col] + iv * gv;
      const float hn = ov * tanhf(cn);
      c_s[row * D_ + col] = cn;
      h_s[row * D_ + col] = hn;
      ot[row * D_ + col]  = hn;
    }
    __syncthreads();
  }
}

// ---------------------------------------------------------------------------
// Phase C: writer scan. 32 blocks x 1024 threads; mem distributed in LDS.
//   Block w owns l in [w*8, w*8+8) for all b: 32*8*128 f32 = 128KB slice.
//   Per step: logits (LDS dots) -> barrier -> redundant softmax ->
//   m_rt partial + global f32 atomics -> barrier -> redundant c_t GEMM +
//   LSTM GEMMs (WMMA, weights streamed/prefetched from L2) -> local mem update.
// ---------------------------------------------------------------------------
__global__ void writer_scan_kernel(const float* __restrict__ oseq,
                                   const float* __restrict__ x,
                                   const float* __restrict__ Ww,
                                   const float* __restrict__ Uw,
                                   const float* __restrict__ bw,
                                   const float* __restrict__ Wc,
                                   const float* __restrict__ bc,
                                   float* __restrict__ logits_g,
                                   float* __restrict__ mrt_g,   // [2][32*128]
                                   unsigned* __restrict__ ctrl, // {cnt, gen}
                                   float* __restrict__ out) {
  extern __shared__ float sm[];
  float* memb  = sm;                    // 32*8*128 mem slice
  float* o_s   = memb + B_ * LSL_ * D_; // 32*128
  float* mrt_s = o_s + B_ * D_;         // 32*128
  float* ct_s  = mrt_s + B_ * D_;       // 32*128
  float* h_s   = ct_s + B_ * D_;        // 32*128
  float* c_s   = h_s + B_ * D_;         // 32*128
  float* z_sl  = c_s + B_ * D_;         // 32*8 own softmax slice

  const int tid  = threadIdx.x;
  const int wid  = tid >> 5;
  const int lane = tid & 31;
  const int lm   = lane & 15;
  const int hi   = lane >> 4;
  const int m0   = (wid & 1) * 16;
  const int j0   = (wid >> 1) * 16;
  const int blk  = blockIdx.x;          // 0..31
  unsigned lgen  = 0;

  // Stage mem slice straight from x (x is never written): mem0 = x.
  for (int i4 = tid; i4 < (B_ * LSL_ * D_) / 4; i4 += blockDim.x) {
    const int e = i4 * 4;
    const int b = e >> 10;              // / (8*128)
    const int r = e & 1023;             // li*128 + d
    ((float4*)memb)[i4] =
        *(const float4*)&x[((size_t)b * L_ + blk * LSL_) * D_ + r];
  }
  for (int i = tid; i < B_ * D_; i += blockDim.x) { h_s[i] = 0.f; c_s[i] = 0.f; }
  __syncthreads();

  for (int t = 0; t < L_; ++t) {
    const int buf = t & 1;

    // S0: cache o_t
    const float* ot = oseq + (size_t)t * B_ * D_;
    for (int i = tid; i < B_ * D_; i += blockDim.x) o_s[i] = ot[i];
    __syncthreads();

    // S1: logits for own slice (tid<256); zero next m_rt buf (256<=tid<384);
    //     idle threads prefetch this step's weight streams into WGP$.
    if (tid < B_ * LSL_) {
      const int b  = tid >> 3;
      const int li = tid & 7;
      const float4* mrow = (const float4*)&memb[(b * LSL_ + li) * D_];
      const float4* orow = (const float4*)&o_s[b * D_];
      float s = 0.f;
#pragma unroll 8
      for (int q = 0; q < D_ / 4; ++q) {
        const float4 m = mrow[q];
        const float4 o = orow[q];
        s += m.x * o.x + m.y * o.y + m.z * o.z + m.w * o.w;
      }
      logits_g[b * L_ + blk * LSL_ + li] = s;
    } else if (tid < B_ * LSL_ + 128) {
      mrt_g[buf * B_ * D_ + blk * 128 + (tid - B_ * LSL_)] = 0.f;
    } else if (tid >= 512) {
      const int pi = tid - 512;                       // 0..511
      for (int q = pi; q < 2048; q += 512) {          // Ww: 256KB
        __builtin_prefetch(Ww + (size_t)q * 32, 0, 3);
        __builtin_prefetch(Uw + (size_t)q * 32, 0, 3); // Uw: 256KB
      }
      for (int q = pi; q < 1024; q += 512)            // Wc: 128KB
        __builtin_prefetch(Wc + (size_t)q * 32, 0, 3);
    }

    gbarrier(ctrl, ctrl + 1, lgen);  // logits complete, m_rt buf zeroed

    // S2: redundant softmax; wave `wid` handles batch row b = wid.
    {
      const int b = wid;
      float v[8];
      float mx = -INFINITY;
#pragma unroll
      for (int q = 0; q < 8; ++q) {
        v[q] = logits_g[b * L_ + lane + 32 * q];
        mx = fmaxf(mx, v[q]);
      }
      for (int off = 16; off > 0; off >>= 1)
        mx = fmaxf(mx, __shfl_xor(mx, off, 32));
      float sum = 0.f;
#pragma unroll
      for (int q = 0; q < 8; ++q) { v[q] = expf(v[q] - mx); sum += v[q]; }
      for (int off = 16; off > 0; off >>= 1)
        sum += __shfl_xor(sum, off, 32);
      const float inv = 1.0f / sum;
      if (lane < LSL_) {
        const float lv = logits_g[b * L_ + blk * LSL_ + lane];
        z_sl[b * LSL_ + lane] = expf(lv - mx) * inv;
      }
    }
    __syncthreads();

    // S3: m_rt partial over own slice -> global f32 atomics.
    for (int p = tid; p < B_ * D_; p += blockDim.x) {
      const int b = p >> 7;
      const int d = p & (D_ - 1);
      float s = 0.f;
#pragma unroll
      for (int li = 0; li < LSL_; ++li)
        s += z_sl[b * LSL_ + li] * memb[(b * LSL_ + li) * D_ + d];
      atomicAdd(&mrt_g[buf * B_ * D_ + p], s);
    }

    gbarrier(ctrl, ctrl + 1, lgen);  // m_rt reduction complete

    // S4: pull full m_rt into LDS.
    for (int i = tid; i < B_ * D_; i += blockDim.x)
      mrt_s[i] = mrt_g[buf * B_ * D_ + i];
    __syncthreads();

    // S5: c_t = [o_t | m_rt] @ Wc + bc  (waves 0..15, K=256, redundant/blk)
    if (wid < 16) {
      v8f acc;
      const float bias = bc[j0 + lm];
#pragma unroll
      for (int vr = 0; vr < 8; ++vr) acc[vr] = bias;
#pragma unroll 2
      for (int k = 0; k < 2 * D_; k += 4) {
        const int kb = k + 2 * hi;  // even; never straddles the 128 boundary
        const float* arow = (kb < D_) ? &o_s[(m0 + lm) * D_ + kb]
                                      : &mrt_s[(m0 + lm) * D_ + (kb - D_)];
        v2f a = *(const v2f*)arow;
        v2f b;
        b.x = Wc[(size_t)kb * D_ + j0 + lm];
        b.y = Wc[(size_t)(kb + 1) * D_ + j0 + lm];
        acc = wmma_f32_16x16x4(a, b, acc);
      }
#pragma unroll
      for (int vr = 0; vr < 8; ++vr)
        ct_s[(m0 + vr + 8 * hi) * D_ + j0 + lm] = acc[vr];
    }
    __syncthreads();

    // S6: z2 = c_t@Ww + h@Uw + bw; gates; local h/c update (waves 0..15).
    v8f acc[4];
    if (wid < 16) {
#pragma unroll
      for (int g = 0; g < 4; ++g) {
        const float bias = bw[g * D_ + j0 + lm];
#pragma unroll
        for (int vr = 0; vr < 8; ++vr) acc[g][vr] = bias;
      }
#pragma unroll 2
      for (int k = 0; k < D_; k += 4) {
        const int kb = k + 2 * hi;
        v2f a = *(const v2f*)&ct_s[(m0 + lm) * D_ + kb];
#pragma unroll
        for (int g = 0; g < 4; ++g) {
          const int col = g * D_ + j0 + lm;
          v2f b;
          b.x = Ww[(size_t)kb * G_ + col];
          b.y = Ww[(size_t)(kb + 1) * G_ + col];
          acc[g] = wmma_f32_16x16x4(a, b, acc[g]);
        }
      }
#pragma unroll 2
      for (int k = 0; k < D_; k += 4) {
        const int kb = k + 2 * hi;
        v2f a = *(const v2f*)&h_s[(m0 + lm) * D_ + kb];
#pragma unroll
        for (int g = 0; g < 4; ++g) {
          const int col = g * D_ + j0 + lm;
          v2f b;
          b.x = Uw[(size_t)kb * G_ + col];
          b.y = Uw[(size_t)(kb + 1) * G_ + col];
          acc[g] = wmma_f32_16x16x4(a, b, acc[g]);
        }
      }
    }
    __syncthreads();  // all h_s reads done
    if (wid < 16) {
#pragma unroll
      for (int vr = 0; vr < 8; ++vr) {
        const int row = m0 + vr + 8 * hi;
        const int col = j0 + lm;
        const float iv = hard_sigmoid(acc[0][vr]);
        const float fv = hard_sigmoid(acc[1][vr]);
        const float gv = tanhf(acc[2][vr]);
        const float ov = hard_sigmoid(acc[3][vr]);
        const float cn = fv * c_s[row * D_ + col] + iv * gv;
        const float hn = ov * tanhf(cn);
        c_s[row * D_ + col] = cn;
        h_s[row * D_ + col] = hn;
      }
    }
    __syncthreads();

    // S7: own mem slice update: mem = mem*(1-z) + h*z  (pure LDS).
    for (int i4 = tid; i4 < (B_ * LSL_ * D_) / 4; i4 += blockDim.x) {
      const int e  = i4 * 4;
      const int b  = e >> 10;
      const int li = (e >> 7) & (LSL_ - 1);
      const int d  = e & (D_ - 1);
      const float z = z_sl[b * LSL_ + li];
      float4 m = ((float4*)memb)[i4];
      const float4 hv = *(const float4*)&h_s[b * D_ + d];
      m.x = m.x * (1.f - z) + hv.x * z;
      m.y = m.y * (1.f - z) + hv.y * z;
      m.z = m.z * (1.f - z) + hv.z * z;
      m.w = m.w * (1.f - z) + hv.w * z;
      ((float4*)memb)[i4] = m;
    }
    // next S0 writes o_s only; the __syncthreads after S0 orders S7 vs S1.
  }

  if (blk == 0)
    for (int i = tid; i < B_ * D_; i += blockDim.x) out[i] = h_s[i];
}

// ---------------------------------------------------------------------------
extern "C" void kernel_launch(void* const* d_in, const int* in_sizes, int n_in,
                              void* d_out, int out_size, void* d_ws, size_t ws_size,
                              hipStream_t stream) {
  const float* x  = (const float*)d_in[0];
  const float* Wr = (const float*)d_in[1];
  const float* Ur = (const float*)d_in[2];
  const float* br = (const float*)d_in[3];
  const float* Ww = (const float*)d_in[4];
  const float* Uw = (const float*)d_in[5];
  const float* bw = (const float*)d_in[6];
  const float* Wc = (const float*)d_in[7];
  const float* bc = (const float*)d_in[8];
  float* out = (float*)d_out;

  // Workspace (~20MB, L2-resident):
  float*    XW       = (float*)d_ws;                      // 8192*512
  float*    oseq     = XW + (size_t)L_ * B_ * G_;         // 256*32*128
  float*    logits_g = oseq + (size_t)L_ * B_ * D_;       // 32*256
  float*    mrt_g    = logits_g + B_ * L_;                // 2*32*128
  unsigned* ctrl     = (unsigned*)(mrt_g + 2 * B_ * D_);  // {cnt, gen}

  // Zero the cross-block control/reduction state every launch (graph-replay
  // deterministic): logits + both m_rt buffers + barrier counters.
  const size_t zbytes = (size_t)(B_ * L_ + 2 * B_ * D_) * sizeof(float) + 256;
  hipMemsetAsync(logits_g, 0, zbytes, stream);

  // Phase A: chip-wide WMMA GEMM (16384 tiles, 8 waves/block).
  precompute_xw_kernel<<<2048, 256, 0, stream>>>(x, Wr, br, XW);

  // Phase B: reader scan, 288KB dynamic LDS on one WGP.
  const int smem_r = (D_ * G_ + 2 * B_ * D_) * (int)sizeof(float);
  hipFuncSetAttribute((const void*)reader_scan_kernel,
                      hipFuncAttributeMaxDynamicSharedMemorySize, smem_r);
  reader_scan_kernel<<<1, 512, smem_r, stream>>>(XW, Ur, oseq);

  // Phase C: writer scan, 32 blocks (1/WGP), ~209KB dynamic LDS each.
  const int smem_w =
      (B_ * LSL_ * D_ + 5 * B_ * D_ + B_ * LSL_) * (int)sizeof(float);
  hipFuncSetAttribute((const void*)writer_scan_kernel,
                      hipFuncAttributeMaxDynamicSharedMemorySize, smem_w);
  writer_scan_kernel<<<NBLK_, 1024, smem_w, stream>>>(
      oseq, x, Ww, Uw, bw, Wc, bc, logits_g, mrt_g, ctrl, out);
}